// GeneralizedCrossMessage_58806692217331
// MI455X (gfx1250) — compile-verified
//
#include <hip/hip_runtime.h>
#include <math.h>

// ---------------------------------------------------------------------------
// GeneralizedCrossMessage for MI455X (gfx1250, wave32, WMMA).
// Pipeline: proj GEMMs -> fused pair-score WMMA kernel (never materializes
// the (B,S,T,HID) tensor) -> softmax -> ctx GEMM -> fused-concat MLP GEMMs.
// All matrix math through V_WMMA_F32_16X16X4_F32 (fp32 to match reference).
// ---------------------------------------------------------------------------

typedef __attribute__((ext_vector_type(2))) float v2f;
typedef __attribute__((ext_vector_type(8))) float v8f;

#define BB   2
#define SS   1024
#define HH   768
#define DD   24
#define HID  96
#define VIN  2304

__device__ __forceinline__ v8f wmma4(v2f a, v2f b, v8f c) {
  return __builtin_amdgcn_wmma_f32_16x16x4_f32(false, a, false, b, (short)0, c,
                                               false, false);
}

// ---------------------------------------------------------------------------
// Generic fp32 WMMA GEMM: C[M,N] = A[M,K] @ B[K,N]   (row-major, K % 4 == 0)
// Wave computes a 16x64 tile (4 n-subtiles); 8 waves stacked in M -> 128x64, blocks 64 rows apart.
// EPI: 0 = none, 1 = relu(x + bias[n]), 2 = alpha*(x + bias[n])
// ---------------------------------------------------------------------------
template <int EPI>
__global__ __launch_bounds__(256) void wmma_gemm_f32(
    const float* __restrict__ A, const float* __restrict__ B,
    float* __restrict__ C, int M, int N, int K, int lda, int ldb, int ldc,
    const float* __restrict__ bias, const float* __restrict__ alpha_p) {
  const int lane = threadIdx.x & 31;
  const int wave = threadIdx.x >> 5;  // 0..7
  const int l15  = lane & 15;
  const int kl   = lane >> 4;         // 0 or 1
  const int m0   = blockIdx.x * 64 + wave * 16;
  const int n0   = blockIdx.y * 64;
  if (m0 >= M) return;

  v8f acc[4] = {};
  const float* Arow = A + (size_t)(m0 + l15) * lda;

  for (int k = 0; k < K; k += 4) {
    const int ka = k + kl * 2;
    v2f a;
    a.x = Arow[ka];
    a.y = Arow[ka + 1];
#pragma unroll
    for (int s4 = 0; s4 < 4; s4++) {
      const int n = n0 + s4 * 16 + l15;
      v2f b;
      if (n < N) {
        b.x = B[(size_t)ka * ldb + n];
        b.y = B[(size_t)(ka + 1) * ldb + n];
      } else {
        b.x = 0.0f; b.y = 0.0f;
      }
      acc[s4] = wmma4(a, b, acc[s4]);
    }
  }

  float alpha = 1.0f;
  if (EPI == 2) alpha = alpha_p[0];
#pragma unroll
  for (int s4 = 0; s4 < 4; s4++) {
    const int n = n0 + s4 * 16 + l15;
    if (n >= N) continue;
    const float bv = (EPI == 0) ? 0.0f : bias[n];
#pragma unroll
    for (int v = 0; v < 8; v++) {
      const int m = m0 + v + 8 * kl;
      if (m >= M) continue;
      float x = acc[s4][v];
      if (EPI == 1) x = fmaxf(x + bv, 0.0f);
      if (EPI == 2) x = alpha * (x + bv);
      C[(size_t)m * ldc + n] = x;
    }
  }
}

// ---------------------------------------------------------------------------
// Pair-score kernel. One block per (b,s). For each 16-row t-tile a wave forms
// X[t, 0:48] on the fly ( Zj*Zi | |Zj-Zi| ), multiplies by W1cd = W1[48:96]
// via WMMA (K=48, N=96), then fuses +tj+ti+b1, relu, dot-with-W2 and a
// 16-lane cross-reduction to emit logits[b,s,t] directly.
// ---------------------------------------------------------------------------
__global__ __launch_bounds__(256) void score_kernel(
    const float* __restrict__ Zj, const float* __restrict__ Zi,
    const float* __restrict__ tj, const float* __restrict__ ti,
    const float* __restrict__ W1, const float* __restrict__ b1,
    const float* __restrict__ W2, float* __restrict__ logits) {
  const int bs   = blockIdx.x;
  const int b    = bs >> 10;
  const int lane = threadIdx.x & 31;
  const int wave = threadIdx.x >> 5;  // 0..7
  const int l15  = lane & 15;
  const int kl   = lane >> 4;

  const float* Wcd = W1 + 48 * HID;                       // 48 x 96, contiguous
  const float* Zjs = Zj + (size_t)bs * DD;                // 24 floats, this s
  const float* Zib = Zi + (size_t)b * SS * DD;
  const float* tjs = tj + (size_t)bs * HID;
  const float* tib = ti + (size_t)b * SS * HID;
  float*       Lrw = logits + (size_t)bs * SS;

  for (int tt = wave; tt < SS / 16; tt += 8) {
    const int t0 = tt * 16;
    const float* Zit = Zib + (size_t)(t0 + l15) * DD;

    v8f acc[6] = {};
#pragma unroll
    for (int kk = 0; kk < 12; kk++) {
      const int c0 = kk * 4 + kl * 2;           // even, never straddles 24
      const int d0 = (c0 < 24) ? c0 : c0 - 24;
      const float zj0 = Zjs[d0], zj1 = Zjs[d0 + 1];
      const float zi0 = Zit[d0], zi1 = Zit[d0 + 1];
      v2f a;
      if (c0 < 24) { a.x = zj0 * zi0;        a.y = zj1 * zi1; }
      else         { a.x = fabsf(zj0 - zi0); a.y = fabsf(zj1 - zi1); }
#pragma unroll
      for (int s6 = 0; s6 < 6; s6++) {
        const int h = s6 * 16 + l15;
        v2f bb;
        bb.x = Wcd[(size_t)c0 * HID + h];
        bb.y = Wcd[(size_t)(c0 + 1) * HID + h];
        acc[s6] = wmma4(a, bb, acc[s6]);
      }
    }

    // Epilogue: relu(acc + tj + ti + b1) . W2, reduced over h.
    float part[8] = {0.f, 0.f, 0.f, 0.f, 0.f, 0.f, 0.f, 0.f};
#pragma unroll
    for (int s6 = 0; s6 < 6; s6++) {
      const int h   = s6 * 16 + l15;
      const float c0f = tjs[h] + b1[h];
      const float w2  = W2[h];
#pragma unroll
      for (int v = 0; v < 8; v++) {
        const int trow = t0 + v + 8 * kl;
        float x = acc[s6][v] + c0f + tib[(size_t)trow * HID + h];
        x = fmaxf(x, 0.0f);
        part[v] += x * w2;
      }
    }
#pragma unroll
    for (int v = 0; v < 8; v++) {
      float p = part[v];
      p += __shfl_xor(p, 1);
      p += __shfl_xor(p, 2);
      p += __shfl_xor(p, 4);
      p += __shfl_xor(p, 8);
      if (l15 == 0) Lrw[t0 + v + 8 * kl] = p;   // +b2 omitted: softmax-invariant
    }
  }
}

// ---------------------------------------------------------------------------
// Row softmax over t with attention mask. One block per (b,s) row of 1024.
// ---------------------------------------------------------------------------
__global__ __launch_bounds__(256) void softmax_kernel(
    float* __restrict__ logits, const float* __restrict__ mask) {
  const int bs = blockIdx.x;
  const int b  = bs >> 10;
  float* row = logits + (size_t)bs * SS;
  const float* mrow = mask + (size_t)b * SS;
  const int tid = threadIdx.x;
  __shared__ float sred[8];

  float v[4];
  float mx = -3.402823466e38f;
#pragma unroll
  for (int i = 0; i < 4; i++) {
    const int t = tid + i * 256;
    float x = row[t] + (1.0f - mrow[t]) * (-3.402823466e38f);
    v[i] = x;
    mx = fmaxf(mx, x);
  }
#pragma unroll
  for (int off = 16; off >= 1; off >>= 1) mx = fmaxf(mx, __shfl_xor(mx, off));
  if ((tid & 31) == 0) sred[tid >> 5] = mx;
  __syncthreads();
  float bm = sred[0];
#pragma unroll
  for (int w = 1; w < 8; w++) bm = fmaxf(bm, sred[w]);
  __syncthreads();

  float sum = 0.0f;
#pragma unroll
  for (int i = 0; i < 4; i++) { v[i] = __expf(v[i] - bm); sum += v[i]; }
#pragma unroll
  for (int off = 16; off >= 1; off >>= 1) sum += __shfl_xor(sum, off);
  if ((tid & 31) == 0) sred[tid >> 5] = sum;
  __syncthreads();
  float bsum = 0.0f;
#pragma unroll
  for (int w = 0; w < 8; w++) bsum += sred[w];
  const float inv = 1.0f / bsum;
#pragma unroll
  for (int i = 0; i < 4; i++) row[tid + i * 256] = v[i] * inv;
}

// ---------------------------------------------------------------------------
// MLP layer 1 with the [ctx, H_j, ctx*H_j] concat fused into the A-fetch.
// h1[M=2048, N=768] = relu( msg_in[2048 x 2304] @ Wv1 + bv1 )
// ---------------------------------------------------------------------------
__global__ __launch_bounds__(256) void mlp1_gemm(
    const float* __restrict__ ctx, const float* __restrict__ Hj,
    const float* __restrict__ Wv1, const float* __restrict__ bv1,
    float* __restrict__ h1) {
  const int lane = threadIdx.x & 31;
  const int wave = threadIdx.x >> 5;
  const int l15  = lane & 15;
  const int kl   = lane >> 4;
  const int m0   = blockIdx.x * 64 + wave * 16;
  const int n0   = blockIdx.y * 64;

  const float* crow = ctx + (size_t)(m0 + l15) * HH;
  const float* hrow = Hj  + (size_t)(m0 + l15) * HH;

  v8f acc[4] = {};
  for (int k = 0; k < VIN; k += 4) {
    const int ka = k + kl * 2;   // even; segment bounds (768,1536) even too
    v2f a;
    if (ka < HH)            { a.x = crow[ka];            a.y = crow[ka + 1]; }
    else if (ka < 2 * HH)   { a.x = hrow[ka - HH];       a.y = hrow[ka - HH + 1]; }
    else { const int kk = ka - 2 * HH;
           a.x = crow[kk] * hrow[kk];
           a.y = crow[kk + 1] * hrow[kk + 1]; }
#pragma unroll
    for (int s4 = 0; s4 < 4; s4++) {
      const int n = n0 + s4 * 16 + l15;
      v2f b;
      b.x = Wv1[(size_t)ka * HH + n];
      b.y = Wv1[(size_t)(ka + 1) * HH + n];
      acc[s4] = wmma4(a, b, acc[s4]);
    }
  }
#pragma unroll
  for (int s4 = 0; s4 < 4; s4++) {
    const int n = n0 + s4 * 16 + l15;
    const float bv = bv1[n];
#pragma unroll
    for (int v = 0; v < 8; v++) {
      const int m = m0 + v + 8 * kl;
      h1[(size_t)m * HH + n] = fmaxf(acc[s4][v] + bv, 0.0f);
    }
  }
}

// ---------------------------------------------------------------------------
extern "C" void kernel_launch(void* const* d_in, const int* in_sizes, int n_in,
                              void* d_out, int out_size, void* d_ws,
                              size_t ws_size, hipStream_t stream) {
  const float* H_j   = (const float*)d_in[0];
  const float* H_i   = (const float*)d_in[1];
  const float* mask  = (const float*)d_in[2];
  const float* Wpj   = (const float*)d_in[3];
  const float* Wpi   = (const float*)d_in[4];
  const float* W1    = (const float*)d_in[5];
  const float* b1    = (const float*)d_in[6];
  const float* W2    = (const float*)d_in[7];
  const float* Wv1   = (const float*)d_in[9];
  const float* bv1   = (const float*)d_in[10];
  const float* Wv2   = (const float*)d_in[11];
  const float* bv2   = (const float*)d_in[12];
  const float* alpha = (const float*)d_in[13];
  float* out = (float*)d_out;

  // Workspace layout (floats)
  float* ws     = (float*)d_ws;
  float* Zj     = ws;                         //  2048*24
  float* Zi     = Zj + 2048 * DD;             //  2048*24
  float* tj     = Zi + 2048 * DD;             //  2048*96
  float* ti     = tj + 2048 * HID;            //  2048*96
  float* logits = ti + 2048 * HID;            //  2*1024*1024
  float* ctx    = logits + (size_t)BB * SS * SS;  // 2048*768
  float* h1     = ctx + 2048 * HH;            //  2048*768

  const dim3 blk(256);

  // Projections: Z = H @ Wp  (M=2048, N=24, K=768)
  wmma_gemm_f32<0><<<dim3(32, 1), blk, 0, stream>>>(
      H_j, Wpj, Zj, 2048, DD, HH, HH, DD, DD, nullptr, nullptr);
  wmma_gemm_f32<0><<<dim3(32, 1), blk, 0, stream>>>(
      H_i, Wpi, Zi, 2048, DD, HH, HH, DD, DD, nullptr, nullptr);

  // tj = Zj @ W1[0:24],  ti = Zi @ W1[24:48]   (M=2048, N=96, K=24)
  wmma_gemm_f32<0><<<dim3(32, 2), blk, 0, stream>>>(
      Zj, W1, tj, 2048, HID, DD, DD, HID, HID, nullptr, nullptr);
  wmma_gemm_f32<0><<<dim3(32, 2), blk, 0, stream>>>(
      Zi, W1 + DD * HID, ti, 2048, HID, DD, DD, HID, HID, nullptr, nullptr);

  // Fused pair scores -> logits (B,S,T)
  score_kernel<<<dim3(BB * SS), blk, 0, stream>>>(Zj, Zi, tj, ti, W1, b1, W2,
                                                  logits);

  // Softmax over t (in place)
  softmax_kernel<<<dim3(BB * SS), blk, 0, stream>>>(logits, mask);

  // ctx = probs @ H_i  per batch (M=1024, N=768, K=1024)
  for (int b = 0; b < BB; b++) {
    wmma_gemm_f32<0><<<dim3(16, 12), blk, 0, stream>>>(
        logits + (size_t)b * SS * SS, H_i + (size_t)b * SS * HH,
        ctx + (size_t)b * SS * HH, SS, HH, SS, SS, HH, HH, nullptr, nullptr);
  }

  // MLP layer 1 (fused concat), layer 2 with alpha*(x+bias)
  mlp1_gemm<<<dim3(32, 12), blk, 0, stream>>>(ctx, H_j, Wv1, bv1, h1);
  wmma_gemm_f32<2><<<dim3(32, 12), blk, 0, stream>>>(
      h1, Wv2, out, 2048, HH, HH, HH, HH, HH, bv2, alpha);
}